// Decoder_60103772340748
// MI455X (gfx1250) — compile-verified
//
#include <hip/hip_runtime.h>
#include <hip/hip_bf16.h>
#include <math.h>

#define B_   64
#define P_   196
#define E_   2048
#define H_   1024
#define A_   1024
#define EMB_ 512
#define V_   10000
#define L_   24
#define T_   23

typedef float v2f __attribute__((ext_vector_type(2)));
typedef float v8f __attribute__((ext_vector_type(8)));

__device__ __forceinline__ float sigmoidf_(float v) { return 1.f / (1.f + expf(-v)); }

// ---------------------------------------------------------------------------
// Sort batch by caption length (stable, descending), emit order / dec_len /
// row gather map for att1 / sorted tokens.
// ---------------------------------------------------------------------------
__global__ void prep_kernel(const int* __restrict__ y, int* __restrict__ order,
                            int* __restrict__ declen, float* __restrict__ declen_out,
                            int* __restrict__ rowmap, int* __restrict__ toks)
{
    __shared__ int len_s[B_];
    int b = threadIdx.x;
    int cnt = 0;
    for (int l = 0; l < L_; ++l) cnt += (y[b * L_ + l] != 0);
    len_s[b] = cnt;
    __syncthreads();
    int rank = 0;
    for (int j = 0; j < B_; ++j) {
        int lj = len_s[j];
        if (lj > cnt || (lj == cnt && j < b)) rank++;   // stable descending
    }
    order[rank] = b;
    declen[rank] = cnt - 1;
    declen_out[rank] = (float)(cnt - 1);
    __syncthreads();
    int ob = order[b];
    for (int p = 0; p < P_; ++p) rowmap[b * P_ + p] = ob * P_ + p;
    for (int t = 0; t < T_; ++t) toks[b * T_ + t] = y[ob * L_ + t];
}

// ---------------------------------------------------------------------------
// mean_enc[b, e] = mean_p enc[order[b], p, e]
// ---------------------------------------------------------------------------
__global__ void mean_kernel(const float* __restrict__ enc, const int* __restrict__ order,
                            float* __restrict__ meanb)
{
    int b = blockIdx.x;
    int e = blockIdx.y * 256 + threadIdx.x;
    int ob = order[b];
    const float* ep = enc + ((size_t)ob * P_) * (size_t)E_ + e;
    float s = 0.f;
    for (int pp = 0; pp < P_; ++pp) s += ep[(size_t)pp * E_];
    meanb[(size_t)b * E_ + e] = s * (1.f / (float)P_);
}

// ---------------------------------------------------------------------------
// Generic f32 WMMA GEMM: C[M,N] (+)= A[M,K] @ B[K,N] (+ bias) with optional
// row-gather on A, and dec_len row masking (for preds output).
//
// Block = 256 threads = 8 waves; wave tile = 16x64 (4 x v8f accumulators).
// V_WMMA_F32_16X16X4_F32 fragment layout (wave32):
//   A 16x4 : m = lane&15, k = 2*(lane>>4) + vgpr   -> contiguous pair (b64 load)
//   B 4x16 : n = lane&15, k = 2*(lane>>4) + vgpr
//   C 16x16: n = lane&15, m = 8*(lane>>4) + vgpr
//
// All call sites honor: N % 16 == 0, K % 8 == 0, lda even, and all operand
// element counts < 2^31.  B is addressed exclusively through TWO uniform
// scalar bases (Bm, Bm+ldb) plus 32-bit per-lane element offsets so every B
// load lowers to the GVS "global_load_b32 v, voff, s[base] scale_offset" form
// (no per-lane 64-bit pointer math in the hot loop).
// Out-of-range column tiles are clamped to valid addresses (branch-free loads)
// and their garbage accumulators are never stored.
// ---------------------------------------------------------------------------
#define WMMA4(ACC, AF, BF) \
    ACC = __builtin_amdgcn_wmma_f32_16x16x4_f32(false, (AF), false, (BF), (short)0, ACC, false, false)

__global__ void __launch_bounds__(256)
wmma_gemm_f32(const float* __restrict__ A, int lda,
              const int* __restrict__ rowmap,
              const float* __restrict__ Bm, int ldb,
              float* __restrict__ C, size_t ldc,
              const float* __restrict__ bias,
              int M, int N, int K,
              int accum,
              const int* __restrict__ declen, int tstep)
{
    const int lane = threadIdx.x & 31;
    const int w    = threadIdx.x >> 5;
    const int wm   = w >> 2;          // 0..1
    const int wn   = w & 3;           // 0..3
    const int m0   = (blockIdx.x * 2 + wm) * 16;
    const int n0   = (blockIdx.y * 4 + wn) * 64;
    if (m0 >= M) return;              // wave-uniform exit keeps EXEC full

    const int l15 = lane & 15;
    const int klo = (lane >> 4) * 2;  // k sub-offset for this lane half

    const int row  = m0 + l15;
    const int arow = rowmap ? rowmap[row] : row;

    // A: uniform base + 32-bit per-lane offset (8B aligned: lda even).
    unsigned aoff = (unsigned)arow * (unsigned)lda + (unsigned)klo;

    // B: two uniform bases (rows +0 and +1 of the current k-block) and
    // 32-bit per-lane offsets; phase-2 offsets (rows klo+4/klo+5) are their
    // own registers so no 64-bit address math appears in the loop.
    const float* BmL = Bm + (size_t)ldb;
    const unsigned ldbu = (unsigned)ldb;

    const int nmax = N - 1;
    const int n_0 = n0      + l15;
    const int n_1 = n0 + 16 + l15;
    const int n_2 = n0 + 32 + l15;
    const int n_3 = n0 + 48 + l15;
    const unsigned kb = (unsigned)klo * ldbu;
    unsigned o0 = kb + (unsigned)(n_0 < N ? n_0 : nmax);
    unsigned o1 = kb + (unsigned)(n_1 < N ? n_1 : nmax);
    unsigned o2 = kb + (unsigned)(n_2 < N ? n_2 : nmax);
    unsigned o3 = kb + (unsigned)(n_3 < N ? n_3 : nmax);
    unsigned p0 = o0 + 4u * ldbu;
    unsigned p1 = o1 + 4u * ldbu;
    unsigned p2 = o2 + 4u * ldbu;
    unsigned p3 = o3 + 4u * ldbu;
    const unsigned bstep = 8u * ldbu;

    const v8f vz = {0.f, 0.f, 0.f, 0.f, 0.f, 0.f, 0.f, 0.f};
    v8f acc0 = vz, acc1 = vz, acc2 = vz, acc3 = vz;

    for (int k = 0; k < K; k += 8) {
        v2f a0 = *(const v2f*)(A + aoff);
        v2f a1 = *(const v2f*)(A + aoff + 4);

        v2f b0, b1, b2, b3, b4, b5, b6, b7;
        b0.x = Bm[o0]; b0.y = BmL[o0];
        b1.x = Bm[o1]; b1.y = BmL[o1];
        b2.x = Bm[o2]; b2.y = BmL[o2];
        b3.x = Bm[o3]; b3.y = BmL[o3];
        b4.x = Bm[p0]; b4.y = BmL[p0];
        b5.x = Bm[p1]; b5.y = BmL[p1];
        b6.x = Bm[p2]; b6.y = BmL[p2];
        b7.x = Bm[p3]; b7.y = BmL[p3];

        WMMA4(acc0, a0, b0);
        WMMA4(acc1, a0, b1);
        WMMA4(acc2, a0, b2);
        WMMA4(acc3, a0, b3);
        WMMA4(acc0, a1, b4);
        WMMA4(acc1, a1, b5);
        WMMA4(acc2, a1, b6);
        WMMA4(acc3, a1, b7);

        aoff += 8;
        o0 += bstep; o1 += bstep; o2 += bstep; o3 += bstep;
        p0 += bstep; p1 += bstep; p2 += bstep; p3 += bstep;
    }

    const int rbase = m0 + (lane >> 4) * 8;
    auto store_tile = [&](v8f acc, int j) {
        int nj = n0 + 16 * j + l15;
        if (nj >= N) return;                  // uniform per tile (N % 16 == 0)
        float bv = bias ? bias[nj] : 0.f;
        #pragma unroll
        for (int i = 0; i < 8; ++i) {
            int r = rbase + i;
            size_t off = (size_t)r * ldc + (size_t)nj;
            float v = acc[i] + bv;
            if (accum) v += C[off];
            if (declen && tstep >= declen[r]) v = 0.f;
            C[off] = v;
        }
    };
    store_tile(acc0, 0);
    store_tile(acc1, 1);
    store_tile(acc2, 2);
    store_tile(acc3, 3);
}

// ---------------------------------------------------------------------------
// e[b,p] = relu(att1[b,p,:] + att2[b,:]) . w_att + b_att ; alpha = softmax_p(e)
// Also writes masked alphas to output.  One block per b (256 thr, P_=196).
// ---------------------------------------------------------------------------
__global__ void attn_softmax_kernel(const float* __restrict__ att1, const float* __restrict__ att2,
                                    const float* __restrict__ w_att, const float* __restrict__ b_att,
                                    const int* __restrict__ declen, int t,
                                    float* __restrict__ alpha, float* __restrict__ alphas_out)
{
    __shared__ float att2_s[A_];
    __shared__ float red[256];
    int b = blockIdx.x, tid = threadIdx.x;
    for (int i = tid; i < A_; i += 256) att2_s[i] = att2[(size_t)b * A_ + i];
    __syncthreads();

    float e = -INFINITY;
    if (tid < P_) {
        const float* a1 = att1 + ((size_t)b * P_ + tid) * (size_t)A_;
        float s = 0.f;
        for (int i = 0; i < A_; ++i) {
            float v = a1[i] + att2_s[i];
            v = v > 0.f ? v : 0.f;
            s += v * w_att[i];
        }
        e = s + b_att[0];
    }
    red[tid] = e; __syncthreads();
    for (int s = 128; s > 0; s >>= 1) { if (tid < s) red[tid] = fmaxf(red[tid], red[tid + s]); __syncthreads(); }
    float mx = red[0]; __syncthreads();
    float ex = (tid < P_) ? expf(e - mx) : 0.f;
    red[tid] = ex; __syncthreads();
    for (int s = 128; s > 0; s >>= 1) { if (tid < s) red[tid] += red[tid + s]; __syncthreads(); }
    float inv = 1.f / red[0];
    if (tid < P_) {
        float al = ex * inv;
        alpha[(size_t)b * P_ + tid] = al;
        int active = (t < declen[b]);
        alphas_out[((size_t)b * T_ + t) * (size_t)P_ + tid] = active ? al : 0.f;
    }
}

// ---------------------------------------------------------------------------
// context[b,e] = sum_p alpha[b,p] * enc[order[b],p,e]
// ---------------------------------------------------------------------------
__global__ void ctx_kernel(const float* __restrict__ enc, const int* __restrict__ order,
                           const float* __restrict__ alpha, float* __restrict__ ctx)
{
    __shared__ float a_s[P_];
    int b = blockIdx.x;
    int e = blockIdx.y * 256 + threadIdx.x;
    if (threadIdx.x < P_) a_s[threadIdx.x] = alpha[(size_t)b * P_ + threadIdx.x];
    __syncthreads();
    int ob = order[b];
    const float* ep = enc + ((size_t)ob * P_) * (size_t)E_ + e;
    float s = 0.f;
    for (int pp = 0; pp < P_; ++pp) s += a_s[pp] * ep[(size_t)pp * E_];
    ctx[(size_t)b * E_ + e] = s;
}

// ---------------------------------------------------------------------------
// x[b] = [ embed_table[tok], sigmoid(gpre[b,:]) * context[b,:] ]
// ---------------------------------------------------------------------------
__global__ void build_x_kernel(const float* __restrict__ emb_tab, const int* __restrict__ toks, int t,
                               const float* __restrict__ gpre, const float* __restrict__ ctx,
                               float* __restrict__ x)
{
    int b = blockIdx.x, tid = threadIdx.x;
    int tok = toks[b * T_ + t];
    float* xb = x + (size_t)b * (EMB_ + E_);
    for (int i = tid; i < EMB_; i += 256) xb[i] = emb_tab[(size_t)tok * EMB_ + i];
    for (int i = tid; i < E_; i += 256) {
        float g = sigmoidf_(gpre[(size_t)b * E_ + i]);
        xb[EMB_ + i] = g * ctx[(size_t)b * E_ + i];
    }
}

// ---------------------------------------------------------------------------
// LSTM cell elementwise (in-place h, c update; gates order i,f,g,o).
// ---------------------------------------------------------------------------
__global__ void lstm_cell_kernel(const float* __restrict__ gates, float* __restrict__ h,
                                 float* __restrict__ c)
{
    int b = blockIdx.x;
    const float* gb = gates + (size_t)b * 4 * H_;
    for (int j = threadIdx.x; j < H_; j += 256) {
        float ig = gb[j];
        float fg = gb[H_ + j];
        float gg = gb[2 * H_ + j];
        float og = gb[3 * H_ + j];
        float cn = sigmoidf_(fg) * c[(size_t)b * H_ + j] + sigmoidf_(ig) * tanhf(gg);
        float hn = sigmoidf_(og) * tanhf(cn);
        c[(size_t)b * H_ + j] = cn;
        h[(size_t)b * H_ + j] = hn;
    }
}

// ---------------------------------------------------------------------------
extern "C" void kernel_launch(void* const* d_in, const int* in_sizes, int n_in,
                              void* d_out, int out_size, void* d_ws, size_t ws_size,
                              hipStream_t stream)
{
    const float* enc       = (const float*)d_in[0];
    const int*   y         = (const int*)  d_in[1];
    const float* W_enc2att = (const float*)d_in[2];
    const float* W_hid2att = (const float*)d_in[3];
    const float* w_att     = (const float*)d_in[4];
    const float* b_att     = (const float*)d_in[5];
    const float* emb_tab   = (const float*)d_in[6];
    const float* W_ih      = (const float*)d_in[7];
    const float* b_ih      = (const float*)d_in[8];
    const float* W_hh      = (const float*)d_in[9];
    const float* b_hh      = (const float*)d_in[10];
    const float* W_fbeta   = (const float*)d_in[11];
    const float* b_fbeta   = (const float*)d_in[12];
    const float* W_fc      = (const float*)d_in[13];
    const float* b_fc      = (const float*)d_in[14];
    const float* W_inith   = (const float*)d_in[15];
    const float* b_inith   = (const float*)d_in[16];
    const float* W_initc   = (const float*)d_in[17];
    const float* b_initc   = (const float*)d_in[18];

    float* preds_out  = (float*)d_out;                          // B*T*V
    float* alphas_out = preds_out + (size_t)B_ * T_ * V_;       // B*T*P
    float* declen_out = alphas_out + (size_t)B_ * T_ * P_;      // B

    char* p = (char*)d_ws;
    auto take = [&](size_t bytes) -> char* {
        char* r = p;
        p += (bytes + 255) & ~(size_t)255;
        return r;
    };
    int*   order  = (int*)  take((size_t)B_ * 4);
    int*   declen = (int*)  take((size_t)B_ * 4);
    int*   rowmap = (int*)  take((size_t)B_ * P_ * 4);
    int*   toks   = (int*)  take((size_t)B_ * T_ * 4);
    float* meanb  = (float*)take((size_t)B_ * E_ * 4);
    float* h      = (float*)take((size_t)B_ * H_ * 4);
    float* c      = (float*)take((size_t)B_ * H_ * 4);
    float* att2   = (float*)take((size_t)B_ * A_ * 4);
    float* alpha  = (float*)take((size_t)B_ * P_ * 4);
    float* ctx    = (float*)take((size_t)B_ * E_ * 4);
    float* gpre   = (float*)take((size_t)B_ * E_ * 4);
    float* x      = (float*)take((size_t)B_ * (EMB_ + E_) * 4);
    float* gates  = (float*)take((size_t)B_ * 4 * H_ * 4);
    float* att1   = (float*)take((size_t)B_ * P_ * A_ * 4);     // 51.4 MB

    prep_kernel<<<1, B_, 0, stream>>>(y, order, declen, declen_out, rowmap, toks);
    mean_kernel<<<dim3(B_, E_ / 256), 256, 0, stream>>>(enc, order, meanb);

    // h0 = mean @ W_inith + b_inith ; c0 = mean @ W_initc + b_initc
    wmma_gemm_f32<<<dim3(B_ / 32, H_ / 256), 256, 0, stream>>>(
        meanb, E_, nullptr, W_inith, H_, h, H_, b_inith, B_, H_, E_, 0, nullptr, 0);
    wmma_gemm_f32<<<dim3(B_ / 32, H_ / 256), 256, 0, stream>>>(
        meanb, E_, nullptr, W_initc, H_, c, H_, b_initc, B_, H_, E_, 0, nullptr, 0);

    // att1 = enc[order] @ W_enc2att   (12544 x 2048 @ 2048 x 1024)
    wmma_gemm_f32<<<dim3((B_ * P_) / 32, A_ / 256), 256, 0, stream>>>(
        enc, E_, rowmap, W_enc2att, A_, att1, (size_t)A_, nullptr,
        B_ * P_, A_, E_, 0, nullptr, 0);

    for (int t = 0; t < T_; ++t) {
        // att2 = h @ W_hid2att
        wmma_gemm_f32<<<dim3(B_ / 32, A_ / 256), 256, 0, stream>>>(
            h, H_, nullptr, W_hid2att, A_, att2, (size_t)A_, nullptr, B_, A_, H_, 0, nullptr, 0);
        // energies + softmax + masked alpha output
        attn_softmax_kernel<<<B_, 256, 0, stream>>>(att1, att2, w_att, b_att,
                                                    declen, t, alpha, alphas_out);
        // context
        ctx_kernel<<<dim3(B_, E_ / 256), 256, 0, stream>>>(enc, order, alpha, ctx);
        // gate pre-activation = h @ W_fbeta + b_fbeta
        wmma_gemm_f32<<<dim3(B_ / 32, E_ / 256), 256, 0, stream>>>(
            h, H_, nullptr, W_fbeta, E_, gpre, (size_t)E_, b_fbeta, B_, E_, H_, 0, nullptr, 0);
        // x = [emb_t, sigmoid(gpre)*context]
        build_x_kernel<<<B_, 256, 0, stream>>>(emb_tab, toks, t, gpre, ctx, x);
        // gates = x @ W_ih + b_ih ; gates += h @ W_hh + b_hh
        wmma_gemm_f32<<<dim3(B_ / 32, (4 * H_) / 256), 256, 0, stream>>>(
            x, EMB_ + E_, nullptr, W_ih, 4 * H_, gates, (size_t)(4 * H_), b_ih,
            B_, 4 * H_, EMB_ + E_, 0, nullptr, 0);
        wmma_gemm_f32<<<dim3(B_ / 32, (4 * H_) / 256), 256, 0, stream>>>(
            h, H_, nullptr, W_hh, 4 * H_, gates, (size_t)(4 * H_), b_hh,
            B_, 4 * H_, H_, 1, nullptr, 0);
        // LSTM cell (updates h, c in place)
        lstm_cell_kernel<<<B_, 256, 0, stream>>>(gates, h, c);
        // preds[:, t, :] = mask(h_new @ W_fc + b_fc)   written straight to d_out
        wmma_gemm_f32<<<dim3(B_ / 32, (V_ + 255) / 256), 256, 0, stream>>>(
            h, H_, nullptr, W_fc, V_, preds_out + (size_t)t * V_, (size_t)T_ * V_, b_fc,
            B_, V_, H_, 0, declen, t);
    }
}